// MaxExtractor_54657753808935
// MI455X (gfx1250) — compile-verified
//
#include <hip/hip_runtime.h>

#define B_  32
#define P_  2048
#define G_  128
#define TPB 1024   // 32 wave32 waves per workgroup

__global__ __launch_bounds__(TPB)
void iou_max_kernel(const float* __restrict__ pred_boxes,
                    const float* __restrict__ pred_scores,
                    const int*   __restrict__ pred_classes,
                    const float* __restrict__ gt_boxes,
                    float* __restrict__ out)
{
    __shared__ float s_gt[G_ * 4];     // staged gt boxes (2 KB)
    __shared__ float s_area[G_];       // gt area, +inf for invalid gt (mask folded in)
    __shared__ float s_riou[TPB / 32];
    __shared__ float s_rsc[TPB / 32];
    __shared__ int   s_rh[TPB / 32];

    const int b   = blockIdx.x;
    const int tid = threadIdx.x;

    // ---- Stage gt boxes into LDS via CDNA5 async-to-LDS DMA (ASYNCcnt path) ----
    if (tid < G_) {
        unsigned lds_off = (unsigned)(unsigned long long)(const void*)(&s_gt[tid * 4]);
        unsigned long long gaddr =
            (unsigned long long)(const void*)(gt_boxes + ((size_t)b * G_ + tid) * 4);
        asm volatile("global_load_async_to_lds_b128 %0, %1, off"
                     :: "v"(lds_off), "v"(gaddr)
                     : "memory");
    }
    asm volatile("s_wait_asynccnt 0" ::: "memory");
    __syncthreads();

    // Per-gt area with validity folded in: invalid gt -> +inf area -> uni=inf ->
    // rcp(inf)=0 -> iou contribution exactly 0 (matches the reference mask).
    if (tid < G_) {
        float gx0 = s_gt[tid * 4 + 0], gy0 = s_gt[tid * 4 + 1];
        float gx1 = s_gt[tid * 4 + 2], gy1 = s_gt[tid * 4 + 3];
        float area  = (gx1 - gx0) * (gy1 - gy0);
        bool  valid = (gx0 + gy0 + gx1 + gy1) != 0.0f;
        s_area[tid] = valid ? area : __builtin_inff();
    }
    __syncthreads();

    const float4* pb4 = (const float4*)(pred_boxes + (size_t)b * P_ * 4);
    const float*  sc  = pred_scores  + (size_t)b * P_;
    const int*    cls = pred_classes + (size_t)b * P_;

    float lmax_iou = 0.0f;    // matches jnp.max of matrix containing zeros
    float lmax_sc  = -1.0f;   // matches where(valid_p, score, -1.0)
    int   lhas     = 0;

    for (int p = tid; p < P_; p += TPB) {
        __builtin_prefetch(&pb4[p], 0, 0);   // global_prefetch_b8
        if (cls[p] == 0) {
            lhas = 1;
            lmax_sc = fmaxf(lmax_sc, sc[p]);
            const float4 pb = pb4[p];
            const float  ap = (pb.z - pb.x) * (pb.w - pb.y);
#pragma unroll 4
            for (int g = 0; g < G_; ++g) {
                float gx0 = s_gt[g * 4 + 0], gy0 = s_gt[g * 4 + 1];
                float gx1 = s_gt[g * 4 + 2], gy1 = s_gt[g * 4 + 3];
                float iw = fminf(gx1, pb.z) - fmaxf(gx0, pb.x);
                float ih = fminf(gy1, pb.w) - fmaxf(gy0, pb.y);
                iw = fmaxf(iw, 0.0f);
                ih = fmaxf(ih, 0.0f);
                float inters = iw * ih;
                float uni = ap + s_area[g] - inters;
                uni = (uni == 0.0f) ? 1.0f : uni;            // safe_uni
                // v_rcp_f32 (TRANS, co-executes with VALU) instead of IEEE div
                float iou = inters * __builtin_amdgcn_rcpf(uni);
                lmax_iou = fmaxf(lmax_iou, iou);
            }
        }
    }

    // ---- wave32 tree reduction (compile-time width 32) ----
#pragma unroll
    for (int off = 16; off > 0; off >>= 1) {
        lmax_iou = fmaxf(lmax_iou, __shfl_xor(lmax_iou, off, 32));
        lmax_sc  = fmaxf(lmax_sc,  __shfl_xor(lmax_sc,  off, 32));
        lhas    |= __shfl_xor(lhas, off, 32);
    }

    const int wave = tid >> 5;
    if ((tid & 31) == 0) {
        s_riou[wave] = lmax_iou;
        s_rsc[wave]  = lmax_sc;
        s_rh[wave]   = lhas;
    }
    __syncthreads();

    // ---- final reduction over the 32 wave partials: exactly one wave ----
    if (tid < 32) {
        float a = s_riou[tid];
        float s = s_rsc[tid];
        int   h = s_rh[tid];
#pragma unroll
        for (int off = 16; off > 0; off >>= 1) {
            a = fmaxf(a, __shfl_xor(a, off, 32));
            s = fmaxf(s, __shfl_xor(s, off, 32));
            h |= __shfl_xor(h, off, 32);
        }
        if (tid == 0) {
            const bool has = (h != 0);
            out[b]      = has ? sqrtf(s + 1.0f) : 0.0f;   // max_prob
            out[B_ + b] = has ? a : 0.0f;                  // max_iou
        }
    }
}

extern "C" void kernel_launch(void* const* d_in, const int* in_sizes, int n_in,
                              void* d_out, int out_size, void* d_ws, size_t ws_size,
                              hipStream_t stream)
{
    const float* pred_boxes   = (const float*)d_in[0];
    const float* pred_scores  = (const float*)d_in[1];
    const int*   pred_classes = (const int*)d_in[2];
    const float* gt_boxes     = (const float*)d_in[3];
    float*       out          = (float*)d_out;

    iou_max_kernel<<<B_, TPB, 0, stream>>>(pred_boxes, pred_scores, pred_classes,
                                           gt_boxes, out);
}